// GCN_13357348290805
// MI455X (gfx1250) — compile-verified
//
#include <hip/hip_runtime.h>
#include <hip/hip_bf16.h>
#include <math.h>

// Problem constants from the reference
#define BATCH   8
#define WINDOW  5
#define NN      20000
#define EPB     320000          // edges per batch graph
#define HID     32
#define KPAD    8               // window padded to a multiple of 4 for WMMA
#define NTOT    (BATCH * NN)    // 160000 batched nodes
#define ETOT    (BATCH * EPB)   // 2560000 batched edges

typedef float v2f __attribute__((ext_vector_type(2)));
typedef float v8f __attribute__((ext_vector_type(8)));

static __device__ __forceinline__ void edge_nodes(const int* __restrict__ ei,
                                                  int e, int& s, int& d) {
    // edge_index is (B, 2, E) row-major; add per-batch node offset.
    int b  = e / EPB;
    int el = e - b * EPB;
    const int* base = ei + (size_t)b * 2 * EPB;
    s = base[el] + b * NN;
    d = base[EPB + el] + b * NN;
}

// n4 float4s; p 16B-aligned (true for all our buffers).
__global__ void zero_f32x4(float4* __restrict__ p, int n4) {
    int i = blockIdx.x * blockDim.x + threadIdx.x;
    if (i < n4) p[i] = make_float4(0.f, 0.f, 0.f, 0.f);
}

// Xp (N x KPAD) = X (N x WINDOW) zero-padded per row.
__global__ void pad_rows_kernel(const float* __restrict__ x, float* __restrict__ xp) {
    int i = blockIdx.x * blockDim.x + threadIdx.x;
    if (i >= NTOT) return;
    const float* __restrict__ r = x + (size_t)i * WINDOW;
    float* __restrict__ o = xp + (size_t)i * KPAD;
    float4 lo = make_float4(r[0], r[1], r[2], r[3]);
    float4 hi = make_float4(r[4], 0.f, 0.f, 0.f);
    reinterpret_cast<float4*>(o)[0] = lo;
    reinterpret_cast<float4*>(o)[1] = hi;
}

__global__ void degree_kernel(const int* __restrict__ ei, float* __restrict__ deg) {
    int e = blockIdx.x * blockDim.x + threadIdx.x;
    if (e >= ETOT) return;
    int s, d; edge_nodes(ei, e, s, d);
    atomicAdd(&deg[d], 1.0f);
}

__global__ void invsqrt_kernel(float* __restrict__ deg) {
    int i = blockIdx.x * blockDim.x + threadIdx.x;
    if (i >= NTOT) return;
    float v = deg[i];
    deg[i] = (v > 0.0f) ? rsqrtf(v) : 0.0f;
}

__global__ void norm_kernel(const int* __restrict__ ei, const float* __restrict__ dis,
                            float* __restrict__ nrm) {
    int e = blockIdx.x * blockDim.x + threadIdx.x;
    if (e >= ETOT) return;
    int s, d; edge_nodes(ei, e, s, d);
    nrm[e] = dis[s] * dis[d];
}

// agg[dst, 0:F] += h[src, 0:F] * norm[e]; rows have stride S (padding stays 0).
// ONE thread per edge: indices/norm loaded once, b128 gathers, fp32 atomics.
template <int F, int S>
__global__ void aggregate_kernel(const int* __restrict__ ei, const float* __restrict__ nrm,
                                 const float* __restrict__ hin, float* __restrict__ agg) {
    int e = blockIdx.x * blockDim.x + threadIdx.x;
    if (e >= ETOT) return;
    int s, d; edge_nodes(ei, e, s, d);
    float w = nrm[e];
    const float* __restrict__ hs = hin + (size_t)s * S;
    float* __restrict__ ad = agg + (size_t)d * S;
    if constexpr ((F % 4) == 0) {
        #pragma unroll
        for (int g = 0; g < F / 4; ++g) {
            float4 v = reinterpret_cast<const float4*>(hs)[g];
            atomicAdd(&ad[4 * g + 0], v.x * w);
            atomicAdd(&ad[4 * g + 1], v.y * w);
            atomicAdd(&ad[4 * g + 2], v.z * w);
            atomicAdd(&ad[4 * g + 3], v.w * w);
        }
    } else {
        #pragma unroll
        for (int f = 0; f < F; ++f) atomicAdd(&ad[f], hs[f] * w);
    }
}

// hout = tanh( hin @ W[0] + agg @ W[1] + bias ); hin/agg: (N,K) with K%4==0,
// W: (2,KW,32) in memory, zero-padded to (2,K,32) in LDS. One wave32 computes a
// 16x32 output tile with V_WMMA_F32_16X16X4_F32; branch-free inner loop so EXEC
// stays all-1s (WMMA requirement) and loads clause cleanly.
// A (16x4 f32): lanes 0-15 -> M=lane, {K=k0,k0+1}; lanes 16-31 -> {K=k0+2,k0+3}.
// B (4x16 f32): lanes 0-15 -> N=lane rows k0,k0+1; lanes 16-31 -> rows k0+2,k0+3.
// C/D (16x16 f32): VGPR v, lanes 0-15 -> M=v, lanes 16-31 -> M=v+8, N=lane&15.
template <int K, int KW>
__global__ void tag_gemm_kernel(const float* __restrict__ hin,
                                const float* __restrict__ agg,
                                const float* __restrict__ W,
                                const float* __restrict__ bias,
                                float* __restrict__ hout, int N) {
    static_assert((K % 4) == 0, "padded K must be a multiple of 4");
    __shared__ float sW[2 * K * HID];
    __shared__ float sB[HID];
    for (int i = threadIdx.x; i < 2 * K * HID; i += blockDim.x) {
        int s   = i / (K * HID);
        int rem = i - s * (K * HID);
        int kw  = rem / HID;
        int col = rem - kw * HID;
        sW[i] = (kw < KW) ? W[((size_t)s * KW + kw) * HID + col] : 0.0f;
    }
    if (threadIdx.x < HID) sB[threadIdx.x] = bias[threadIdx.x];
    __syncthreads();

    int lane = threadIdx.x & 31;
    int wave = (blockIdx.x * blockDim.x + threadIdx.x) >> 5;
    int row0 = wave * 16;
    if (row0 >= N) return;                 // wave-uniform guard

    const int m     = lane & 15;
    const int khalf = (lane >> 4) * 2;     // 0 for lanes 0-15, 2 for lanes 16-31

    v8f acc0 = {0.f,0.f,0.f,0.f,0.f,0.f,0.f,0.f};
    v8f acc1 = {0.f,0.f,0.f,0.f,0.f,0.f,0.f,0.f};

    #pragma unroll
    for (int s = 0; s < 2; ++s) {
        const float* __restrict__ A   = s ? agg : hin;
        const float* __restrict__ sWk = sW + s * K * HID;
        #pragma unroll
        for (int c = 0; c < K / 4; ++c) {
            const int k0 = c * 4 + khalf;
            v2f a = *reinterpret_cast<const v2f*>(&A[(size_t)(row0 + m) * K + k0]);
            v2f b0, b1;
            b0.x = sWk[k0 * HID + m];
            b0.y = sWk[(k0 + 1) * HID + m];
            b1.x = sWk[k0 * HID + 16 + m];
            b1.y = sWk[(k0 + 1) * HID + 16 + m];
            acc0 = __builtin_amdgcn_wmma_f32_16x16x4_f32(
                       false, a, false, b0, (short)0, acc0, false, false);
            acc1 = __builtin_amdgcn_wmma_f32_16x16x4_f32(
                       false, a, false, b1, (short)0, acc1, false, false);
        }
    }

    const int rbase = row0 + ((lane >> 4) ? 8 : 0);
    #pragma unroll
    for (int v = 0; v < 8; ++v) {
        int r = rbase + v;
        float x0 = acc0[v] + sB[m];
        float x1 = acc1[v] + sB[16 + m];
        hout[(size_t)r * HID + m]      = tanhf(x0);
        hout[(size_t)r * HID + 16 + m] = tanhf(x1);
    }
}

__global__ void readout_kernel(const float* __restrict__ h, const float* __restrict__ Wreg,
                               const float* __restrict__ breg, float* __restrict__ out) {
    int i = blockIdx.x * blockDim.x + threadIdx.x;
    if (i >= NTOT) return;
    float acc = breg[0];
    const float4* hr = reinterpret_cast<const float4*>(h + (size_t)i * HID);
    #pragma unroll
    for (int g = 0; g < HID / 4; ++g) {
        float4 v = hr[g];
        acc += v.x * Wreg[4 * g + 0] + v.y * Wreg[4 * g + 1]
             + v.z * Wreg[4 * g + 2] + v.w * Wreg[4 * g + 3];
    }
    out[i] = acc;
}

static inline int cdiv(long long a, long long b) { return (int)((a + b - 1) / b); }

extern "C" void kernel_launch(void* const* d_in, const int* in_sizes, int n_in,
                              void* d_out, int out_size, void* d_ws, size_t ws_size,
                              hipStream_t stream) {
    const float* X    = (const float*)d_in[0];   // (B,WIN,NN) == (NTOT,5) after reshape
    const int*   ei   = (const int*)  d_in[1];   // (B,2,E) int32
    const float* W0   = (const float*)d_in[2];   // (2,5,32)
    const float* b0   = (const float*)d_in[3];
    const float* W1   = (const float*)d_in[4];   // (2,32,32)
    const float* b1   = (const float*)d_in[5];
    const float* W2   = (const float*)d_in[6];   // (2,32,32)
    const float* b2   = (const float*)d_in[7];
    const float* Wreg = (const float*)d_in[8];   // (32,1)
    const float* breg = (const float*)d_in[9];

    float* out    = (float*)d_out;               // first output: (B,NN) = NTOT
    float* hfinal = out + NTOT;                  // second output: (NTOT,32)

    // Workspace layout (~77.4 MB); all offsets 128B-aligned.
    float* dis = (float*)d_ws;                   // NTOT      (deg -> 1/sqrt(deg))
    float* nrm = dis + NTOT;                     // ETOT
    float* agg = nrm + ETOT;                     // NTOT*HID (layer 0 uses NTOT*KPAD of it)
    float* hA  = agg + (size_t)NTOT * HID;       // NTOT*HID
    float* hB  = hA  + (size_t)NTOT * HID;       // NTOT*HID
    float* Xp  = hB  + (size_t)NTOT * HID;       // NTOT*KPAD (zero-padded X)

    const int T = 256;

    // Normalization: deg -> rsqrt -> per-edge norm
    zero_f32x4<<<cdiv(NTOT / 4, T), T, 0, stream>>>((float4*)dis, NTOT / 4);
    degree_kernel<<<cdiv(ETOT, T), T, 0, stream>>>(ei, dis);
    invsqrt_kernel<<<cdiv(NTOT, T), T, 0, stream>>>(dis);
    norm_kernel<<<cdiv(ETOT, T), T, 0, stream>>>(ei, dis, nrm);

    const int gemm_blocks = cdiv((long long)(NTOT / 16) * 32, T);  // 1 wave per 16-row tile
    const int agg_blocks  = cdiv(ETOT, T);

    // Layer 0: true K = WINDOW = 5, padded to KPAD = 8
    pad_rows_kernel<<<cdiv(NTOT, T), T, 0, stream>>>(X, Xp);
    zero_f32x4<<<cdiv((long long)NTOT * KPAD / 4, T), T, 0, stream>>>((float4*)agg, NTOT * KPAD / 4);
    aggregate_kernel<WINDOW, KPAD><<<agg_blocks, T, 0, stream>>>(ei, nrm, Xp, agg);
    tag_gemm_kernel<KPAD, WINDOW><<<gemm_blocks, T, 0, stream>>>(Xp, agg, W0, b0, hA, NTOT);

    // Layer 1: K = 32
    zero_f32x4<<<cdiv((long long)NTOT * HID / 4, T), T, 0, stream>>>((float4*)agg, NTOT * HID / 4);
    aggregate_kernel<HID, HID><<<agg_blocks, T, 0, stream>>>(ei, nrm, hA, agg);
    tag_gemm_kernel<HID, HID><<<gemm_blocks, T, 0, stream>>>(hA, agg, W1, b1, hB, NTOT);

    // Layer 2: K = 32, write h directly into its output slot
    zero_f32x4<<<cdiv((long long)NTOT * HID / 4, T), T, 0, stream>>>((float4*)agg, NTOT * HID / 4);
    aggregate_kernel<HID, HID><<<agg_blocks, T, 0, stream>>>(ei, nrm, hB, agg);
    tag_gemm_kernel<HID, HID><<<gemm_blocks, T, 0, stream>>>(hB, agg, W2, b2, hfinal, NTOT);

    // Readout: out = h @ Wreg + breg
    readout_kernel<<<cdiv(NTOT, T), T, 0, stream>>>(hfinal, Wreg, breg, out);
}